// RevGNNLayer_48747878810305
// MI455X (gfx1250) — compile-verified
//
#include <hip/hip_runtime.h>

typedef float v2f __attribute__((ext_vector_type(2)));
typedef float v8f __attribute__((ext_vector_type(8)));

#define LN_EPS 1e-5f

// Packed-transposed weight layout: for K-pair p = k/2 (k even),
//   WtP[p * WT_PITCH + j*2 + (k&1)] = W[j][k]   (i.e. Wt[k][j])
// One WMMA B fragment {Wt[kk][j], Wt[kk+1][j]} is then a single b64 load.
// WT_PITCH = 256 + 32 pad dwords: 288 % 64 == 32, so consecutive K-pairs
// start 32 banks apart -> the two half-waves (K-pairs p and p+1) of a
// ds_load_b64 hit disjoint bank halves. Conflict-free.
#define WT_PITCH  288
#define WT_FLOATS (64 * WT_PITCH)   // 18432 floats = 73728 B per matrix

// ---------------------------------------------------------------------------
// Kernel 1: LayerNorm + ReLU + dropout mask -> h ; also zero agg and cnt.
// One wave32 per row (128 floats = 32 lanes x float4). 8 rows / 256-thr block.
// ---------------------------------------------------------------------------
__global__ void __launch_bounds__(256)
ln_relu_mask_zero(const float* __restrict__ x,
                  const float* __restrict__ mask,
                  const float* __restrict__ gamma,
                  const float* __restrict__ beta,
                  float* __restrict__ h,
                  float* __restrict__ agg,
                  float* __restrict__ cnt,
                  int N) {
    const int wave = threadIdx.x >> 5;
    const int lane = threadIdx.x & 31;
    const int row  = blockIdx.x * 8 + wave;
    if (row >= N) return;

    const float4 v = ((const float4*)(x + (size_t)row * 128))[lane];
    float s  = v.x + v.y + v.z + v.w;
    float ss = v.x * v.x + v.y * v.y + v.z * v.z + v.w * v.w;
#pragma unroll
    for (int m = 16; m >= 1; m >>= 1) {
        s  += __shfl_xor(s,  m, 32);
        ss += __shfl_xor(ss, m, 32);
    }
    const float mu   = s * (1.0f / 128.0f);
    const float var  = ss * (1.0f / 128.0f) - mu * mu;
    const float rstd = rsqrtf(var + LN_EPS);

    const float4 mk = ((const float4*)(mask + (size_t)row * 128))[lane];
    const float4 g  = ((const float4*)gamma)[lane];
    const float4 b  = ((const float4*)beta)[lane];

    float4 o;
    o.x = fmaxf((v.x - mu) * rstd * g.x + b.x, 0.0f) * mk.x;
    o.y = fmaxf((v.y - mu) * rstd * g.y + b.y, 0.0f) * mk.y;
    o.z = fmaxf((v.z - mu) * rstd * g.z + b.z, 0.0f) * mk.z;
    o.w = fmaxf((v.w - mu) * rstd * g.w + b.w, 0.0f) * mk.w;

    ((float4*)(h + (size_t)row * 128))[lane] = o;

    const float4 z = make_float4(0.0f, 0.0f, 0.0f, 0.0f);
    ((float4*)(agg + (size_t)row * 128))[lane] = z;
    if (lane == 0) cnt[row] = 0.0f;
}

// ---------------------------------------------------------------------------
// Kernel 2: pack W_l, W_r (row-major [j][k]) into the transposed,
// pair-interleaved, bank-padded layout described above.
// ---------------------------------------------------------------------------
__global__ void __launch_bounds__(256)
pack_w(const float* __restrict__ Wl,
       const float* __restrict__ Wr,
       float* __restrict__ wtl_p,
       float* __restrict__ wtr_p) {
    const int idx = blockIdx.x * blockDim.x + threadIdx.x;   // 0 .. 16383
    if (idx >= 128 * 128) return;
    const int j = idx >> 7;
    const int k = idx & 127;
    const int dst = (k >> 1) * WT_PITCH + j * 2 + (k & 1);
    wtl_p[dst] = Wl[idx];
    wtr_p[dst] = Wr[idx];
}

// ---------------------------------------------------------------------------
// Kernel 3: edge scatter (mean-agg numerator + degree count).
// One wave32 per edge: coalesced float4 row read, f32 atomics into agg.
// ---------------------------------------------------------------------------
__global__ void __launch_bounds__(256)
scatter_mean(const long long* __restrict__ ei,   // [2, E] int64
             const float* __restrict__ h,
             float* __restrict__ agg,
             float* __restrict__ cnt,
             int E) {
    const int wave = threadIdx.x >> 5;
    const int lane = threadIdx.x & 31;
    const int e    = blockIdx.x * 8 + wave;
    if (e >= E) return;

    const long long src = ei[e];
    const long long dst = ei[(size_t)E + e];

    const float4 v = ((const float4*)(h + (size_t)src * 128))[lane];
    float* ad = agg + (size_t)dst * 128 + lane * 4;
    atomicAdd(ad + 0, v.x);
    atomicAdd(ad + 1, v.y);
    atomicAdd(ad + 2, v.z);
    atomicAdd(ad + 3, v.w);
    if (lane == 0) atomicAdd(cnt + dst, 1.0f);
}

// ---------------------------------------------------------------------------
// Kernel 4: out = (agg/max(cnt,1)) @ W_l^T + b_l + h @ W_r^T via
// V_WMMA_F32_16X16X4_F32, with both packed weight matrices staged into
// 144 KB of CDNA5 LDS (320 KB/WGP) so all 8 waves per block read B
// fragments as conflict-free ds_load_b64 instead of hammering L1.
//
// One wave computes a 16-row x 128-col slab: 8 column tiles in v8f
// accumulators, K-loop of 32 steps, 2 WMMAs per tile per step.
//
// Fragment layouts (ISA 7.12.2, wave32):
//   A 16x4 f32 (2 VGPRs): lanes 0-15 -> row M=lane,    K={k,k+1};
//                         lanes 16-31 -> row M=lane-16, K={k+2,k+3}.
//   B 4x16  f32 (2 VGPRs): mirrored -> lane half selects K offset 0 / 2.
//   C/D 16x16 f32 (8 VGPRs): VGPR r -> row (r + 8*half), col = lane&15.
// ---------------------------------------------------------------------------
__global__ void __launch_bounds__(256)
sage_gemm(const float* __restrict__ agg,
          const float* __restrict__ cnt,
          const float* __restrict__ h,
          const float* __restrict__ wtl_p,   // packed W_l^T
          const float* __restrict__ wtr_p,   // packed W_r^T
          const float* __restrict__ bias,
          float* __restrict__ out,
          int N) {
    __shared__ float smem[2 * WT_FLOATS];    // 147456 B of LDS
    float* __restrict__ sl = smem;
    float* __restrict__ sr = smem + WT_FLOATS;

    // Cooperative verbatim copy-in of the pre-packed weight images
    // (coalesced b64 global loads -> ds stores).
    {
        const float2* __restrict__ gl = (const float2*)wtl_p;
        const float2* __restrict__ gr = (const float2*)wtr_p;
        float2* __restrict__ dl = (float2*)sl;
        float2* __restrict__ dr = (float2*)sr;
        for (int i = threadIdx.x; i < WT_FLOATS / 2; i += 256) {
            dl[i] = gl[i];
            dr[i] = gr[i];
        }
    }
    __syncthreads();

    const int wave    = threadIdx.x >> 5;
    const int lane    = threadIdx.x & 31;
    const int rowBase = (blockIdx.x * 8 + wave) * 16;
    if (rowBase >= N) return;      // wave-uniform: EXEC all-ones for WMMA

    const int half = lane >> 4;    // which half-wave
    const int l16  = lane & 15;
    const int aRow = rowBase + l16;

    const float rs = 1.0f / fmaxf(cnt[aRow], 1.0f);
    const float* __restrict__ ar = agg + (size_t)aRow * 128;
    const float* __restrict__ hr = h   + (size_t)aRow * 128;

    v8f acc[8];
#pragma unroll
    for (int t = 0; t < 8; ++t) {
        const float bv = bias[t * 16 + l16];
        acc[t] = (v8f){bv, bv, bv, bv, bv, bv, bv, bv};
    }

    for (int k = 0; k < 128; k += 4) {
        const int kk = k + 2 * half;          // this lane-half's K offset
        const int kp = kk >> 1;               // K-pair row in packed layout
        const v2f Aa = {ar[kk] * rs, ar[kk + 1] * rs};   // mean_agg fragment
        const v2f Ah = {hr[kk],      hr[kk + 1]};        // h fragment

        const float* __restrict__ bl_base = sl + (size_t)kp * WT_PITCH + l16 * 2;
        const float* __restrict__ br_base = sr + (size_t)kp * WT_PITCH + l16 * 2;
#pragma unroll
        for (int t = 0; t < 8; ++t) {
            const float2 bl2 = *(const float2*)(bl_base + t * 32);  // ds_load_b64
            const v2f Bl = {bl2.x, bl2.y};
            acc[t] = __builtin_amdgcn_wmma_f32_16x16x4_f32(
                false, Aa, false, Bl, (short)0, acc[t], false, false);

            const float2 br2 = *(const float2*)(br_base + t * 32);  // ds_load_b64
            const v2f Br = {br2.x, br2.y};
            acc[t] = __builtin_amdgcn_wmma_f32_16x16x4_f32(
                false, Ah, false, Br, (short)0, acc[t], false, false);
        }
    }

#pragma unroll
    for (int t = 0; t < 8; ++t) {
#pragma unroll
        for (int r = 0; r < 8; ++r) {
            const int row = rowBase + r + 8 * half;
            out[(size_t)row * 128 + t * 16 + l16] = acc[t][r];
        }
    }
}

// ---------------------------------------------------------------------------
// Launch
// ---------------------------------------------------------------------------
extern "C" void kernel_launch(void* const* d_in, const int* in_sizes, int n_in,
                              void* d_out, int out_size, void* d_ws, size_t ws_size,
                              hipStream_t stream) {
    const float*     x     = (const float*)d_in[0];
    const long long* ei    = (const long long*)d_in[1];   // int64 edge_index
    const float*     mask  = (const float*)d_in[2];
    const float*     gamma = (const float*)d_in[3];
    const float*     beta  = (const float*)d_in[4];
    const float*     Wl    = (const float*)d_in[5];
    const float*     bl    = (const float*)d_in[6];
    const float*     Wr    = (const float*)d_in[7];

    const int N = in_sizes[0] / 128;
    const int E = in_sizes[1] / 2;

    // Workspace (floats): h[N*128] | agg[N*128] | cnt[N] | WtL_p | WtR_p
    float* ws    = (float*)d_ws;
    float* h     = ws;
    float* agg   = h   + (size_t)N * 128;
    float* cnt   = agg + (size_t)N * 128;
    float* wtl_p = cnt + (size_t)N;
    float* wtr_p = wtl_p + WT_FLOATS;

    float* out = (float*)d_out;

    ln_relu_mask_zero<<<(N + 7) / 8, 256, 0, stream>>>(x, mask, gamma, beta,
                                                       h, agg, cnt, N);
    pack_w<<<(128 * 128 + 255) / 256, 256, 0, stream>>>(Wl, Wr, wtl_p, wtr_p);
    scatter_mean<<<(E + 7) / 8, 256, 0, stream>>>(ei, h, agg, cnt, E);
    sage_gemm<<<(N + 127) / 128, 256, 0, stream>>>(agg, cnt, h, wtl_p, wtr_p,
                                                   bl, out, N);
}